// DMPNNLayer_30777735643631
// MI455X (gfx1250) — compile-verified
//
#include <hip/hip_runtime.h>
#include <hip/hip_bf16.h>

// DMPNN layer for MI455X (gfx1250, wave32, WMMA + async global->LDS).
// B=32, N=256, H=128, E=16.
// Roofline: ~145 MB traffic (128 MB edge_attr) vs ~1.1 GFLOP -> HBM-bound,
// ~6 us floor at 23.3 TB/s. So: e_sum is an async-copy streaming reduction
// (ASYNCcnt-tracked, NT hint, 4-deep LDS pipeline), and all GEMM contractions
// use V_WMMA_F32_16X16X4_F32 at full f32 precision (compute is free here; a
// bf16 repack would cost accuracy for zero wall-clock gain).

typedef __attribute__((ext_vector_type(2))) float v2f;
typedef __attribute__((ext_vector_type(8))) float v8f;

#define B_ 32
#define N_ 256
#define H_ 128
#define E_ 16

static __device__ __forceinline__ v8f wmma_f32(v2f a, v2f b, v8f c) {
  // 8 args: (neg_a, A, neg_b, B, c_mod, C, reuse_a, reuse_b)
  return __builtin_amdgcn_wmma_f32_16x16x4_f32(
      false, a, false, b, (short)0, c, false, false);
}

// gate[i,j] = any_b(adj[b,i,j] != 0)
__global__ void k_gate(const int* __restrict__ adj, float* __restrict__ gate) {
  const int t = blockIdx.x * blockDim.x + threadIdx.x;  // 0 .. N*N-1
  int s = 0;
#pragma unroll
  for (int b = 0; b < B_; ++b) s |= adj[b * (N_ * N_) + t];
  gate[t] = s ? 1.0f : 0.0f;
}

// count[j] = sum_i gate[i,j]
__global__ void k_count(const float* __restrict__ gate, float* __restrict__ count) {
  const int j = threadIdx.x;  // 256 threads, 1 block
  float s = 0.0f;
  for (int i = 0; i < N_; ++i) s += gate[i * N_ + j];
  count[j] = s;
}

// e_sum[b,j,e] = sum_i gate[i,j] * edge_attr[b,i,j,e]
// One block per (b, 16-wide j tile). The 128 MB edge_attr stream goes through
// the CDNA5 async global->LDS path: per 4-row (4 KB) chunk each of the 8 waves
// issues one GLOBAL_LOAD_ASYNC_TO_LDS_B128 (32 lanes x 16 B), 4 chunks in
// flight (ASYNCcnt), NT temporal hint (read-once data).
__global__ void k_esum(const float* __restrict__ gate,
                       const float* __restrict__ edge,
                       float* __restrict__ esum) {
  // Single shared block => LDS offset 0, so float index*4 is the raw LDS byte
  // address the async VDST operand needs.
  // [0,4096)    : gate tile  gtile[i][jl], 16 KB
  // [4096,8192) : 4 buffers x 4 rows x 256 floats, 16 KB
  __shared__ float smem[8192];
  const int b  = blockIdx.x >> 4;
  const int j0 = (blockIdx.x & 15) * 16;
  const int t  = threadIdx.x;  // 0..255

  {  // stage gate tile (coalesced 64B rows)
    const int jl = t & 15, ib = t >> 4;
#pragma unroll
    for (int k = 0; k < 16; ++k) {
      const int i = ib + 16 * k;
      smem[i * 16 + jl] = gate[i * N_ + j0 + jl];
    }
  }

  // uniform base for this block's (b, j0) panel; per-row stride N_*E_ floats
  const float* sbase = edge + (((size_t)b * N_) * N_ + j0) * E_;
  const int rr  = t >> 6;   // row within a 4-row chunk (4 x 64 lanes)
  const int col = t & 63;   // 16-byte unit within the 1 KB row
  const unsigned vbase = (unsigned)(rr * (N_ * E_ * 4) + col * 16);
  const unsigned lbase = (unsigned)(4096 * 4 + rr * 1024 + col * 16);

  // prologue: fill the 4-deep async pipeline
#pragma unroll
  for (int c = 0; c < 4; ++c) {
    const unsigned voff = vbase + (unsigned)(c * 4 * N_ * E_ * 4);
    const unsigned loff = lbase + (unsigned)(c * 4096);
    asm volatile("global_load_async_to_lds_b128 %0, %1, %2 th:TH_LOAD_NT"
                 :: "v"(loff), "v"(voff), "s"(sbase) : "memory");
  }

  const int jl = t >> 4;  // consumer j within tile; e = t & 15
  float acc = 0.0f;
  for (int c = 0; c < N_ / 4; ++c) {
    // oldest of <=4 outstanding per-wave async ops (chunk c) is complete
    asm volatile("s_wait_asynccnt 0x3" ::: "memory");
    __syncthreads();  // all waves' pieces of chunk c visible (also gate tile)
    const int bufo = 4096 + (c & 3) * 1024;
    const int i0 = c * 4;
#pragma unroll
    for (int r = 0; r < 4; ++r)
      acc += smem[(i0 + r) * 16 + jl] * smem[bufo + r * 256 + t];
    __syncthreads();  // everyone done reading this buffer before refill
    if (c + 4 < N_ / 4) {
      const unsigned voff = vbase + (unsigned)((c + 4) * 4 * N_ * E_ * 4);
      const unsigned loff = lbase + (unsigned)((c & 3) * 4096);
      asm volatile("global_load_async_to_lds_b128 %0, %1, %2 th:TH_LOAD_NT"
                   :: "v"(loff), "v"(voff), "s"(sbase) : "memory");
    }
  }
  esum[((size_t)b * N_ + j0) * E_ + t] = acc;
}

// gh[b,j,o] = sum_i gate[i,j] * h[b,i,o]   (per-batch gate^T @ h, WMMA f32)
// One wave = one 16x16 tile. Tile id: b*(16*8) + jt*8 + ot.
__global__ void k_gh(const float* __restrict__ gate,
                     const float* __restrict__ h,
                     float* __restrict__ gh) {
  const int lane = threadIdx.x & 31;
  const int tile = blockIdx.x * (blockDim.x >> 5) + (threadIdx.x >> 5);
  const int b  = tile >> 7;
  const int jt = (tile >> 3) & 15;
  const int ot = tile & 7;
  const int m  = lane & 15;   // M (A) / N (B,C)
  const int hh = lane >> 4;   // lane half -> K pair
  const int j0 = jt * 16, o0 = ot * 16;
  const float* __restrict__ hb = h + (size_t)b * N_ * H_;
  v8f c = {};
#pragma unroll 4
  for (int k = 0; k < N_; k += 4) {
    const int ka = k + 2 * hh;
    v2f a, bb;
    a[0]  = gate[(ka + 0) * N_ + j0 + m];   // A[M=j][K=i] = gate[i][j]
    a[1]  = gate[(ka + 1) * N_ + j0 + m];
    bb[0] = hb[(ka + 0) * H_ + o0 + m];     // B[K=i][N=o] = h[b,i,o]
    bb[1] = hb[(ka + 1) * H_ + o0 + m];
    c = wmma_f32(a, bb, c);
  }
  float* __restrict__ out = gh + ((size_t)b * N_ + j0) * H_ + o0;
#pragma unroll
  for (int r = 0; r < 8; ++r) out[(r + 8 * hh) * H_ + m] = c[r];
}

// y[b,j,o] = h[b,j,o] + mask(j<nn[b]) * ( gh[b,j,:]·Wh[o,:] + e_sum[b,j,:]·We[o,:]
//                                         + count[j]*W_b[o] )
__global__ void k_msg(const float* __restrict__ gh,
                      const float* __restrict__ esum,
                      const float* __restrict__ count,
                      const float* __restrict__ Ww,   // H x (H+E) row-major
                      const float* __restrict__ Wb,   // H
                      const float* __restrict__ h,
                      const int* __restrict__ num_nodes,
                      float* __restrict__ y) {
  const int lane = threadIdx.x & 31;
  const int tile = blockIdx.x * (blockDim.x >> 5) + (threadIdx.x >> 5);
  const int b  = tile >> 7;
  const int jt = (tile >> 3) & 15;
  const int ot = tile & 7;
  const int m  = lane & 15;
  const int hh = lane >> 4;
  const int j0 = jt * 16, o0 = ot * 16;
  const int LDW = H_ + E_;  // 144
  const float* __restrict__ arow = gh + ((size_t)b * N_ + j0 + m) * H_;
  const float* __restrict__ erow = esum + ((size_t)b * N_ + j0 + m) * E_;
  const float* __restrict__ wrow = Ww + (o0 + m) * LDW;
  v8f c = {};
#pragma unroll 4
  for (int k = 0; k < H_; k += 4) {          // gh @ Wh^T
    const int ka = k + 2 * hh;
    const v2f a  = *reinterpret_cast<const v2f*>(arow + ka);  // b64 load
    const v2f bb = *reinterpret_cast<const v2f*>(wrow + ka);  // B[K][o]=Wh[o][K]
    c = wmma_f32(a, bb, c);
  }
#pragma unroll
  for (int k = 0; k < E_; k += 4) {          // e_sum @ We^T
    const int ka = k + 2 * hh;
    const v2f a  = *reinterpret_cast<const v2f*>(erow + ka);
    const v2f bb = *reinterpret_cast<const v2f*>(wrow + H_ + ka);
    c = wmma_f32(a, bb, c);
  }
  const int nn = num_nodes[b];
  const float wb = Wb[o0 + m];
#pragma unroll
  for (int r = 0; r < 8; ++r) {
    const int row = j0 + r + 8 * hh;
    float msg = c[r] + count[row] * wb;
    msg = (row < nn) ? msg : 0.0f;
    const size_t idx = ((size_t)b * N_ + row) * H_ + o0 + m;
    y[idx] = h[idx] + msg;
  }
}

// h_new[b,j,o] = y[b,j,:] · U_w[o,:] + U_b[o]
__global__ void k_out(const float* __restrict__ y,
                      const float* __restrict__ Uw,   // H x H row-major
                      const float* __restrict__ Ub,
                      float* __restrict__ out) {
  const int lane = threadIdx.x & 31;
  const int tile = blockIdx.x * (blockDim.x >> 5) + (threadIdx.x >> 5);
  const int b  = tile >> 7;
  const int jt = (tile >> 3) & 15;
  const int ot = tile & 7;
  const int m  = lane & 15;
  const int hh = lane >> 4;
  const int j0 = jt * 16, o0 = ot * 16;
  const float* __restrict__ arow = y + ((size_t)b * N_ + j0 + m) * H_;
  const float* __restrict__ urow = Uw + (o0 + m) * H_;
  const float ub = Ub[o0 + m];
  v8f c;
#pragma unroll
  for (int r = 0; r < 8; ++r) c[r] = ub;    // bias folded into accumulator
#pragma unroll 4
  for (int k = 0; k < H_; k += 4) {
    const int ka = k + 2 * hh;
    const v2f a  = *reinterpret_cast<const v2f*>(arow + ka);
    const v2f bb = *reinterpret_cast<const v2f*>(urow + ka);  // B[K][o]=U_w[o][K]
    c = wmma_f32(a, bb, c);
  }
#pragma unroll
  for (int r = 0; r < 8; ++r)
    out[((size_t)b * N_ + j0 + r + 8 * hh) * H_ + o0 + m] = c[r];
}

extern "C" void kernel_launch(void* const* d_in, const int* in_sizes, int n_in,
                              void* d_out, int out_size, void* d_ws, size_t ws_size,
                              hipStream_t stream) {
  const float* h    = (const float*)d_in[0];
  const float* edge = (const float*)d_in[1];
  const int*   adj  = (const int*)d_in[2];
  const int*   nn   = (const int*)d_in[3];
  const float* Ww   = (const float*)d_in[4];
  const float* Wb   = (const float*)d_in[5];
  const float* Uw   = (const float*)d_in[6];
  const float* Ub   = (const float*)d_in[7];
  float* out = (float*)d_out;

  float* ws    = (float*)d_ws;
  float* gate  = ws;                               // N*N          = 65536
  float* count = gate + (N_ * N_);                 // N            = 256
  float* esum  = count + N_;                       // B*N*E        = 131072
  float* gh    = esum + (size_t)B_ * N_ * E_;      // B*N*H        = 1048576
  float* y     = gh + (size_t)B_ * N_ * H_;        // B*N*H        = 1048576
  (void)in_sizes; (void)n_in; (void)out_size; (void)ws_size;

  // 1. gate + count (8 MB adj read, trivial)
  k_gate<<<(N_ * N_) / 256, 256, 0, stream>>>(adj, gate);
  k_count<<<1, N_, 0, stream>>>(gate, count);

  // 2. e_sum: async-copy streams the 128 MB edge_attr once (the roofline
  //    bottleneck; 4-deep ASYNCcnt pipeline, NT hint)
  k_esum<<<B_ * (N_ / 16), 256, 0, stream>>>(gate, edge, esum);

  // 3-5. WMMA GEMMs: 4096 16x16 tiles each, 4 waves (128 thr) per block
  k_gh <<<1024, 128, 0, stream>>>(gate, h, gh);
  k_msg<<<1024, 128, 0, stream>>>(gh, esum, count, Ww, Wb, h, nn, y);
  k_out<<<1024, 128, 0, stream>>>(y, Uw, Ub, out);
}